// Decoder_22952305229940
// MI455X (gfx1250) — compile-verified
//
#include <hip/hip_runtime.h>

// Bidirectional 5-layer LSTM decoder, T=200 feedback steps, B=1024, H=32.
// One persistent kernel; each workgroup owns 16 batch rows for the entire
// recurrence (rows are independent end-to-end). All state lives in LDS.
// Matrix math: V_WMMA_F32_16X16X4_F32 (full fp32 fidelity vs reference).
// Whh B-fragments / layer-0 Wih / biases live in registers; accumulators
// start at zero (inline-0 SRC2) and bias is added at the gate store so no
// broadcast vectors get hoisted+spilled.

#define Hh   32
#define Ll   5
#define INw  3
#define OUTw 3
#define Tt   200
#define Bb   1024

typedef __attribute__((ext_vector_type(2))) float v2f;
typedef __attribute__((ext_vector_type(8))) float v8f;

// Branch-free activations: one v_exp_f32 + v_rcp_f32 each, no exec-mask dance.
__device__ __forceinline__ float sigf(float x) {
  return __builtin_amdgcn_rcpf(1.0f + __expf(-x));
}
__device__ __forceinline__ float tanh_f(float x) {
  x = fminf(fmaxf(x, -15.0f), 15.0f);          // keep e^{2x} finite (v_med3)
  float e = __expf(2.0f * x);
  return 1.0f - 2.0f * __builtin_amdgcn_rcpf(e + 1.0f);
}

__device__ __forceinline__ v8f wmma4(v8f acc, v2f a, v2f b) {
  // D = A(16x4,f32) * B(4x16,f32) + C
  return __builtin_amdgcn_wmma_f32_16x16x4_f32(false, a, false, b, (short)0, acc, false, false);
}

__global__ __launch_bounds__(512) void lstm_decoder_kernel(
    const float* __restrict__ h0,   const float* __restrict__ c0,
    const float* __restrict__ start_token,
    const float* __restrict__ W_ih0, const float* __restrict__ W_hh0,
    const float* __restrict__ b_ih0, const float* __restrict__ b_hh0,
    const float* __restrict__ W_ih,  const float* __restrict__ W_hh,
    const float* __restrict__ b_ih,  const float* __restrict__ b_hh,
    const float* __restrict__ W_lin, const float* __restrict__ b_lin,
    float* __restrict__ out)
{
  // Per-workgroup persistent state (16 batch rows):
  __shared__ float s_h[2 * Ll][16][Hh];      // 20 KB hidden state, idx = 2*l + dir
  __shared__ float s_c[2 * Ll][16][Hh];      // 20 KB cell state
  __shared__ float s_gates[2][16][4 * Hh];   // 16 KB gate pre-activations (i|f|g|o)
  __shared__ float s_x[16][2 * Hh];          //  4 KB layer input (bidir concat)
  __shared__ float s_y[16][4];               // fed-back output, col 3 = zero pad
  __shared__ float s_wlin[OUTw][2 * Hh];     // head weights (hot in LDS)
  __shared__ float s_blin[OUTw];

  const int tid  = threadIdx.x;
  const int lane = tid & 31;
  const int wave = tid >> 5;      // 16 waves
  const int d    = wave >> 3;     // direction 0/1
  const int nt   = wave & 7;      // 16-column gate tile within 4H=128
  const int base = blockIdx.x * 16;

  // ---- init LDS state from h0/c0 (2L,B,H), start_token, head weights ----
  for (int e = tid; e < 2 * Ll * 16 * Hh; e += 512) {
    int idx = e >> 9; int rem = e & 511; int mm = rem >> 5; int j = rem & 31;
    size_t g = ((size_t)idx * Bb + base + mm) * Hh + j;
    s_h[idx][mm][j] = h0[g];
    s_c[idx][mm][j] = c0[g];
  }
  if (tid < 64) { int mm = tid >> 2, j = tid & 3; s_y[mm][j] = (j < INw) ? start_token[j] : 0.0f; }
  if (tid < OUTw * 2 * Hh) s_wlin[tid / (2 * Hh)][tid % (2 * Hh)] = W_lin[tid];
  if (tid < OUTw) s_blin[tid] = b_lin[tid];

  // WMMA fragment coordinates (ISA 7.12.2, wave32):
  //   A: lane l, vgpr v -> A[l%16][2*(l/16)+v]
  //   B: lane l, vgpr v -> B[2*(l/16)+v][l%16]
  //   C/D: lane l, vgpr r -> C[r + 8*(l/16)][l%16]
  const int m    = lane & 15;
  const int hi2  = (lane >> 4) << 1;
  const int coln = nt * 16 + m;             // gate column in [0,128)
  const int rowo = (lane >> 4) << 3;        // C/D row offset

  // ---- preload loop-invariant B-fragments + biases into registers ----
  v2f wih0f;                                // layer-0 Wih (K=3 zero-padded to 4)
  {
    const float* Wid = W_ih0 + d * 128 * INw;
    wih0f[0] = (hi2     < INw) ? Wid[coln * INw + hi2]     : 0.0f;
    wih0f[1] = (hi2 + 1 < INw) ? Wid[coln * INw + hi2 + 1] : 0.0f;
  }
  v2f whhf[Ll][8];                          // Whh fragments, all layers (80 VGPRs)
  float biasv[Ll];                          // scalar per layer; added at store
  #pragma unroll
  for (int l = 0; l < Ll; ++l) {
    const float* Whd = (l == 0) ? (W_hh0 + d * 128 * Hh)
                                : (W_hh + (size_t)((l - 1) * 2 + d) * 128 * Hh);
    #pragma unroll
    for (int k0 = 0; k0 < 8; ++k0)
      whhf[l][k0] = *(const v2f*)&Whd[coln * Hh + k0 * 4 + hi2];
    if (l == 0) biasv[0] = b_ih0[d * 128 + coln] + b_hh0[d * 128 + coln];
    else {
      int li = (l - 1) * 2 + d;
      biasv[l] = b_ih[li * 128 + coln] + b_hh[li * 128 + coln];
    }
  }

  for (int t = 0; t < Tt; ++t) {
    #pragma unroll
    for (int l = 0; l < Ll; ++l) {
      __syncthreads();  // inputs (s_x or s_y) and s_gates free for this layer

      v8f acc = {};     // zero accumulator -> backend can use inline-0 SRC2;
                        // nothing loop-invariant to hoist/spill

      if (l == 0) {
        // x @ Wih^T with K=3 zero-padded to 4 (s_y[:,3] == 0)
        v2f a;
        a[0] = s_y[m][hi2];
        a[1] = s_y[m][hi2 + 1];
        acc = wmma4(acc, a, wih0f);
      } else {
        // x @ Wih^T, K = 64 (B-fragments L0/L2-hot global loads)
        const float* Wid = W_ih + (size_t)((l - 1) * 2 + d) * 128 * (2 * Hh);
        #pragma unroll
        for (int k0 = 0; k0 < 2 * Hh; k0 += 4) {
          int kk = k0 + hi2;
          v2f a = *(const v2f*)&s_x[m][kk];                 // 8B-aligned ds_load
          v2f b = *(const v2f*)&Wid[coln * (2 * Hh) + kk];
          acc = wmma4(acc, a, b);
        }
      }
      {
        // h @ Whh^T, K = 32 (weights already in registers)
        #pragma unroll
        for (int k0 = 0; k0 < 8; ++k0) {
          v2f a = *(const v2f*)&s_h[2 * l + d][m][k0 * 4 + hi2];
          acc = wmma4(acc, a, whhf[l][k0]);
        }
      }
      // bias added here: 8 dual-issueable adds off the WMMA dependency chain
      #pragma unroll
      for (int r = 0; r < 8; ++r)
        s_gates[d][r + rowo][coln] = acc[r] + biasv[l];

      __syncthreads();  // gates visible; s_x/s_h reads done -> safe to overwrite

      // ---- LSTM cell elementwise: 2 dir * 16 rows * 32 hidden = 1024 elems ----
      #pragma unroll
      for (int q = 0; q < 2; ++q) {
        int e   = tid + q * 512;
        int dir = e >> 9, rem = e & 511, mm = rem >> 5, j = rem & 31;
        float gi = s_gates[dir][mm][j];
        float gf = s_gates[dir][mm][j + 32];
        float gg = s_gates[dir][mm][j + 64];
        float go = s_gates[dir][mm][j + 96];
        float cv = s_c[2 * l + dir][mm][j];
        float cn = sigf(gf) * cv + sigf(gi) * tanh_f(gg);
        float hn = sigf(go) * tanh_f(cn);
        s_c[2 * l + dir][mm][j] = cn;
        s_h[2 * l + dir][mm][j] = hn;
        s_x[mm][dir * Hh + j] = hn;   // concat(fwd, rev) -> next layer input
      }
    }

    __syncthreads();  // s_x (layer-4 output) ready for the head

    // ---- y = tanh(x @ W_lin^T + b_lin), feed back + emit ----
    if (tid < 16 * OUTw) {
      int mm = tid / OUTw, o = tid % OUTw;
      float a2 = s_blin[o];
      #pragma unroll
      for (int k = 0; k < 2 * Hh; ++k) a2 += s_x[mm][k] * s_wlin[o][k];
      float y = tanh_f(a2);
      out[((size_t)t * Bb + base + mm) * OUTw + o] = y;
      s_y[mm][o] = y;                // pad column s_y[mm][3] stays 0 forever
    }
    // next iteration's top-of-layer __syncthreads orders s_y write vs. reads
  }
}

extern "C" void kernel_launch(void* const* d_in, const int* in_sizes, int n_in,
                              void* d_out, int out_size, void* d_ws, size_t ws_size,
                              hipStream_t stream) {
  (void)in_sizes; (void)n_in; (void)d_ws; (void)ws_size; (void)out_size;
  const float* h0          = (const float*)d_in[0];
  const float* c0          = (const float*)d_in[1];
  const float* start_token = (const float*)d_in[2];
  const float* W_ih0       = (const float*)d_in[3];
  const float* W_hh0       = (const float*)d_in[4];
  const float* b_ih0       = (const float*)d_in[5];
  const float* b_hh0       = (const float*)d_in[6];
  const float* W_ih        = (const float*)d_in[7];
  const float* W_hh        = (const float*)d_in[8];
  const float* b_ih        = (const float*)d_in[9];
  const float* b_hh        = (const float*)d_in[10];
  const float* W_lin       = (const float*)d_in[11];
  const float* b_lin       = (const float*)d_in[12];
  float* out = (float*)d_out;

  dim3 grid(Bb / 16);   // 64 workgroups, each owns 16 batch rows end-to-end
  dim3 block(512);      // 16 wave32s: one 16x16 gate tile per wave
  lstm_decoder_kernel<<<grid, block, 0, stream>>>(
      h0, c0, start_token, W_ih0, W_hh0, b_ih0, b_hh0,
      W_ih, W_hh, b_ih, b_hh, W_lin, b_lin, out);
}